// GATEncoder_49744311222479
// MI455X (gfx1250) — compile-verified
//
#include <hip/hip_runtime.h>

#define IN_F  128
#define HID_F 64
#define NEG_SLOPE 0.2f
#define LN_EPS 1e-5f

typedef float v2f __attribute__((ext_vector_type(2)));
typedef float v8f __attribute__((ext_vector_type(8)));

__device__ __forceinline__ float elu_f(float x)   { return x > 0.f ? x : expm1f(x); }
__device__ __forceinline__ float lrelu_f(float x) { return x > 0.f ? x : NEG_SLOPE * x; }

// order-preserving float<->uint mapping so unsigned atomicMax == float max
__device__ __forceinline__ unsigned ord_enc(float f) {
  unsigned u = __float_as_uint(f);
  return ((int)u < 0) ? ~u : (u | 0x80000000u);
}
__device__ __forceinline__ float ord_dec(unsigned o) {
  unsigned u = (o & 0x80000000u) ? (o & 0x7FFFFFFFu) : ~o;
  return __uint_as_float(u);
}

// ---------------- input GEMM: h = elu(x @ W_in + b_in), fp32 WMMA ------------
__global__ void k_gemm_in(const float* __restrict__ x, const float* __restrict__ W,
                          const float* __restrict__ b, float* __restrict__ h, int nTiles) {
  int wave = (blockIdx.x * blockDim.x + threadIdx.x) >> 5;
  if (wave >= nTiles) return;                  // wave-uniform: EXEC stays all-1s
  int lane = threadIdx.x & 31;
  int half = lane >> 4;                        // K sub-pair select (ISA A layout)
  int l15  = lane & 15;
  int row  = wave * 16 + l15;

  v8f acc0 = {}, acc1 = {}, acc2 = {}, acc3 = {};
  const float* xrow = x + (size_t)row * IN_F;
#pragma unroll 4
  for (int k0 = 0; k0 < IN_F; k0 += 4) {
    int ka = k0 + 2 * half;
    v2f a; a.x = xrow[ka]; a.y = xrow[ka + 1];
    const float* w0 = W + (size_t)ka * HID_F;
    const float* w1 = w0 + HID_F;
    v2f b0, b1, b2, b3;
    b0.x = w0[l15];      b0.y = w1[l15];
    b1.x = w0[16 + l15]; b1.y = w1[16 + l15];
    b2.x = w0[32 + l15]; b2.y = w1[32 + l15];
    b3.x = w0[48 + l15]; b3.y = w1[48 + l15];
    acc0 = __builtin_amdgcn_wmma_f32_16x16x4_f32(false, a, false, b0, (short)0, acc0, false, false);
    acc1 = __builtin_amdgcn_wmma_f32_16x16x4_f32(false, a, false, b1, (short)0, acc1, false, false);
    acc2 = __builtin_amdgcn_wmma_f32_16x16x4_f32(false, a, false, b2, (short)0, acc2, false, false);
    acc3 = __builtin_amdgcn_wmma_f32_16x16x4_f32(false, a, false, b3, (short)0, acc3, false, false);
  }
  int rbase = wave * 16 + 8 * half;            // C layout: vgpr r -> M=r or 8+r
#pragma unroll
  for (int r = 0; r < 8; ++r) {
    float* hr = h + (size_t)(rbase + r) * HID_F;
    hr[l15]      = elu_f(acc0[r] + b[l15]);
    hr[16 + l15] = elu_f(acc1[r] + b[16 + l15]);
    hr[32 + l15] = elu_f(acc2[r] + b[32 + l15]);
    hr[48 + l15] = elu_f(acc3[r] + b[48 + l15]);
  }
}

// ------------- per-layer dual GEMM: xl = h@Wl+bl, xr = h@Wr+br ---------------
__global__ void k_gemm_dual(const float* __restrict__ h,
                            const float* __restrict__ Wl, const float* __restrict__ bl,
                            const float* __restrict__ Wr, const float* __restrict__ br,
                            float* __restrict__ xl, float* __restrict__ xr, int nTiles) {
  int wave = (blockIdx.x * blockDim.x + threadIdx.x) >> 5;
  if (wave >= nTiles) return;
  int lane = threadIdx.x & 31;
  int half = lane >> 4;
  int l15  = lane & 15;
  int row  = wave * 16 + l15;

  v8f al0 = {}, al1 = {}, al2 = {}, al3 = {};
  v8f ar0 = {}, ar1 = {}, ar2 = {}, ar3 = {};
  const float* hrow = h + (size_t)row * HID_F;
#pragma unroll 4
  for (int k0 = 0; k0 < HID_F; k0 += 4) {
    int ka = k0 + 2 * half;
    v2f a; a.x = hrow[ka]; a.y = hrow[ka + 1];
    const float* wl0 = Wl + (size_t)ka * HID_F;  const float* wl1 = wl0 + HID_F;
    const float* wr0 = Wr + (size_t)ka * HID_F;  const float* wr1 = wr0 + HID_F;
    v2f b0, b1, b2, b3;
    b0.x = wl0[l15];      b0.y = wl1[l15];
    b1.x = wl0[16 + l15]; b1.y = wl1[16 + l15];
    b2.x = wl0[32 + l15]; b2.y = wl1[32 + l15];
    b3.x = wl0[48 + l15]; b3.y = wl1[48 + l15];
    al0 = __builtin_amdgcn_wmma_f32_16x16x4_f32(false, a, false, b0, (short)0, al0, false, false);
    al1 = __builtin_amdgcn_wmma_f32_16x16x4_f32(false, a, false, b1, (short)0, al1, false, false);
    al2 = __builtin_amdgcn_wmma_f32_16x16x4_f32(false, a, false, b2, (short)0, al2, false, false);
    al3 = __builtin_amdgcn_wmma_f32_16x16x4_f32(false, a, false, b3, (short)0, al3, false, false);
    b0.x = wr0[l15];      b0.y = wr1[l15];
    b1.x = wr0[16 + l15]; b1.y = wr1[16 + l15];
    b2.x = wr0[32 + l15]; b2.y = wr1[32 + l15];
    b3.x = wr0[48 + l15]; b3.y = wr1[48 + l15];
    ar0 = __builtin_amdgcn_wmma_f32_16x16x4_f32(false, a, false, b0, (short)0, ar0, false, false);
    ar1 = __builtin_amdgcn_wmma_f32_16x16x4_f32(false, a, false, b1, (short)0, ar1, false, false);
    ar2 = __builtin_amdgcn_wmma_f32_16x16x4_f32(false, a, false, b2, (short)0, ar2, false, false);
    ar3 = __builtin_amdgcn_wmma_f32_16x16x4_f32(false, a, false, b3, (short)0, ar3, false, false);
  }
  int rbase = wave * 16 + 8 * half;
#pragma unroll
  for (int r = 0; r < 8; ++r) {
    float* pl = xl + (size_t)(rbase + r) * HID_F;
    float* pr = xr + (size_t)(rbase + r) * HID_F;
    pl[l15]      = al0[r] + bl[l15];
    pl[16 + l15] = al1[r] + bl[16 + l15];
    pl[32 + l15] = al2[r] + bl[32 + l15];
    pl[48 + l15] = al3[r] + bl[48 + l15];
    pr[l15]      = ar0[r] + br[l15];
    pr[16 + l15] = ar1[r] + br[16 + l15];
    pr[32 + l15] = ar2[r] + br[32 + l15];
    pr[48 + l15] = ar3[r] + br[48 + l15];
  }
}

// ---------------- per-layer init: y=0, m=-inf(ordered), s=0 ------------------
__global__ void k_init(float* __restrict__ y, unsigned* __restrict__ m,
                       float* __restrict__ s, int N_) {
  int t = blockIdx.x * blockDim.x + threadIdx.x;
  if (t < N_ * HID_F) y[t] = 0.f;
  if (t < N_ * 4) { m[t] = 0x007FFFFFu; s[t] = 0.f; }   // ord_enc(-inf)
}

// ------------- edge pass 1: logits + segment max (ordered atomicMax) ---------
__global__ void k_edge_logits(const int* __restrict__ src, const int* __restrict__ dst,
                              const float* __restrict__ xl, const float* __restrict__ xr,
                              const float* __restrict__ att, float* __restrict__ elog,
                              unsigned* __restrict__ mmax, int E_, int H, int D) {
  int t = blockIdx.x * blockDim.x + threadIdx.x;
  if (t >= E_ * H) return;
  int e = t / H, hh = t - e * H;
  int sj = src[e], di = dst[e];
  const float* pl = xl + (size_t)sj * HID_F + hh * D;
  const float* pr = xr + (size_t)di * HID_F + hh * D;
  const float* pa = att + hh * D;
  float lg = 0.f;
  for (int d = 0; d < D; ++d) lg += pa[d] * lrelu_f(pl[d] + pr[d]);
  elog[t] = lg;
  atomicMax(&mmax[di * H + hh], ord_enc(lg));
}

// ------------- edge pass 2: ex = exp(logit - m[dst]); segment sum ------------
__global__ void k_edge_exp(const int* __restrict__ dst, const unsigned* __restrict__ mmax,
                           float* __restrict__ elog, float* __restrict__ s, int E_, int H) {
  int t = blockIdx.x * blockDim.x + threadIdx.x;
  if (t >= E_ * H) return;
  int e = t / H, hh = t - e * H;
  int di = dst[e];
  float ex = expf(elog[t] - ord_dec(mmax[di * H + hh]));
  elog[t] = ex;
  atomicAdd(&s[di * H + hh], ex);
}

// ------------- edge pass 3: y[dst] += (ex/s[dst]) * xl[src] ------------------
__global__ void k_edge_scatter(const int* __restrict__ src, const int* __restrict__ dst,
                               const float* __restrict__ xl, const float* __restrict__ elog,
                               const float* __restrict__ s, float* __restrict__ y,
                               long long total, int H, int shiftD) {
  long long t = (long long)blockIdx.x * blockDim.x + threadIdx.x;
  if (t >= total) return;
  int e   = (int)(t >> 6);
  int col = (int)(t & 63);
  int hh  = col >> shiftD;
  int di  = dst[e];
  float alpha = elog[e * H + hh] / s[di * H + hh];
  atomicAdd(&y[(size_t)di * HID_F + col], alpha * xl[(size_t)src[e] * HID_F + col]);
}

// ------- epilogue: h = elu(LayerNorm(y + bias)*gamma + beta) + h  ------------
__global__ void k_epilogue(float* __restrict__ h, const float* __restrict__ y,
                           const float* __restrict__ bias, const float* __restrict__ gamma,
                           const float* __restrict__ beta, int N_) {
  int node = (blockIdx.x * blockDim.x + threadIdx.x) >> 5;
  if (node >= N_) return;
  int lane = threadIdx.x & 31;
  const float* yr = y + (size_t)node * HID_F;
  float v0 = yr[lane]      + bias[lane];
  float v1 = yr[lane + 32] + bias[lane + 32];
  float sum = v0 + v1;
  for (int off = 16; off > 0; off >>= 1) sum += __shfl_xor(sum, off, 32);
  float mu = sum * (1.f / 64.f);
  float d0 = v0 - mu, d1 = v1 - mu;
  float sq = d0 * d0 + d1 * d1;
  for (int off = 16; off > 0; off >>= 1) sq += __shfl_xor(sq, off, 32);
  float rs = rsqrtf(sq * (1.f / 64.f) + LN_EPS);
  float* hr = h + (size_t)node * HID_F;
  float r0 = hr[lane], r1 = hr[lane + 32];
  hr[lane]      = elu_f(d0 * rs * gamma[lane]      + beta[lane])      + r0;
  hr[lane + 32] = elu_f(d1 * rs * gamma[lane + 32] + beta[lane + 32]) + r1;
}

extern "C" void kernel_launch(void* const* d_in, const int* in_sizes, int n_in,
                              void* d_out, int out_size, void* d_ws, size_t ws_size,
                              hipStream_t stream) {
  (void)n_in; (void)out_size; (void)ws_size;
  const float* x   = (const float*)d_in[0];
  const int*   ei  = (const int*)d_in[1];
  const float* Win = (const float*)d_in[2];
  const float* bin = (const float*)d_in[3];
  const int N_ = in_sizes[0] / IN_F;
  const int E_ = in_sizes[1] / 2;
  const int* src = ei;
  const int* dst = ei + E_;
  float* h = (float*)d_out;                     // h lives in d_out across layers

  float* ws = (float*)d_ws;
  size_t nh = (size_t)N_ * HID_F;
  float*    xl = ws;
  float*    xr = xl + nh;
  float*    yb = xr + nh;
  unsigned* mm = (unsigned*)(yb + nh);
  float*    ss = (float*)(mm + (size_t)N_ * 4);
  float*    el = ss + (size_t)N_ * 4;           // E*H logits / exps

  const int nTiles = (N_ + 15) / 16;
  dim3 blk(256);
  const int gGemm = (nTiles * 32 + 255) / 256;  // 8 waves per block, 1 tile per wave
  k_gemm_in<<<gGemm, blk, 0, stream>>>(x, Win, bin, h, nTiles);

  for (int i = 0; i < 3; ++i) {
    const float* Wl  = (const float*)d_in[4 + 8 * i];
    const float* bl  = (const float*)d_in[5 + 8 * i];
    const float* Wr  = (const float*)d_in[6 + 8 * i];
    const float* br  = (const float*)d_in[7 + 8 * i];
    const float* att = (const float*)d_in[8 + 8 * i];
    const float* bia = (const float*)d_in[9 + 8 * i];
    const float* gam = (const float*)d_in[10 + 8 * i];
    const float* bet = (const float*)d_in[11 + 8 * i];
    const int H = (i == 2) ? 1 : 4;
    const int D = HID_F / H;
    const int shiftD = (i == 2) ? 6 : 4;

    k_gemm_dual<<<gGemm, blk, 0, stream>>>(h, Wl, bl, Wr, br, xl, xr, nTiles);
    const int gInit = (int)((nh + 255) / 256);
    k_init<<<gInit, blk, 0, stream>>>(yb, mm, ss, N_);
    const int gEH = (E_ * H + 255) / 256;
    k_edge_logits<<<gEH, blk, 0, stream>>>(src, dst, xl, xr, att, el, mm, E_, H, D);
    k_edge_exp<<<gEH, blk, 0, stream>>>(dst, mm, el, ss, E_, H);
    const long long tot = (long long)E_ * HID_F;
    const int gSc = (int)((tot + 255) / 256);
    k_edge_scatter<<<gSc, blk, 0, stream>>>(src, dst, xl, el, ss, yb, tot, H, shiftD);
    const int gEp = (N_ * 32 + 255) / 256;
    k_epilogue<<<gEp, blk, 0, stream>>>(h, yb, bia, gam, bet, N_);
  }
}